// SAGE_13237089207003
// MI455X (gfx1250) — compile-verified
//
#include <hip/hip_runtime.h>
#include <math.h>

typedef __attribute__((ext_vector_type(2))) float v2f;
typedef __attribute__((ext_vector_type(8))) float v8f;

#define DIM 64   // NODE_DIM == HIDDEN == 64
#define K2  128  // concat [mean | h] feature width
#define MT  4    // M-tiles per wave in the WMMA GEMMs

// ---------------------------------------------------------------- utilities
__global__ void zero_f32_kernel(float* __restrict__ p, size_t n) {
  size_t i = (size_t)blockIdx.x * blockDim.x + threadIdx.x;
  size_t stride = (size_t)gridDim.x * blockDim.x;
  for (; i < n; i += stride) p[i] = 0.0f;
}

__global__ void gather_kernel(const int* __restrict__ x, const float* __restrict__ emb,
                              float* __restrict__ h0, int nNodes) {
  int gid = blockIdx.x * blockDim.x + threadIdx.x;
  int n = gid >> 6, d = gid & 63;
  if (n >= nNodes) return;
  h0[(size_t)n * DIM + d] = emb[(size_t)x[n] * DIM + d];
}

__global__ void flags_kernel(const int* __restrict__ mask, int* __restrict__ flags, int nMask) {
  int i = blockIdx.x * blockDim.x + threadIdx.x;
  if (i < nMask) flags[mask[i]] = 1;
}

// Pack [Wl;Wr] (K2 x N) into pair-interleaved layout:
// Wpk[((k>>1)*N + n)*2 + (k&1)]  ->  one b64 load yields W[k][n], W[k+1][n].
__global__ void pack_w_kernel(const float* __restrict__ Wl, const float* __restrict__ Wr,
                              float* __restrict__ Wpk, int N) {
  int gid = blockIdx.x * blockDim.x + threadIdx.x;
  if (gid >= K2 * N) return;
  int k = gid / N;
  int n = gid - k * N;
  float v = (k < 64) ? Wl[k * N + n] : Wr[(k - 64) * N + n];
  Wpk[((size_t)(k >> 1) * N + n) * 2 + (k & 1)] = v;
}

// ------------------------------------------------------------- edge scatter
// 64 lanes per edge -> one f32 atomic per (edge,dim). src/dst loads are
// wave-uniform (broadcast). Layer-2 version skips edges whose dst is unmasked.
template <bool FILTER>
__global__ void __launch_bounds__(256)
scatter_kernel(const int* __restrict__ eidx, const float* __restrict__ h,
               float* __restrict__ agg, float* __restrict__ deg,
               const int* __restrict__ flags, int nEdges) {
  long long t = (long long)blockIdx.x * 256 + threadIdx.x;
  int e = (int)(t >> 6);
  int d = (int)(t & 63);
  if (e >= nEdges) return;
  int dst = eidx[nEdges + e];     // edge_index[1][e]
  if (FILTER && !flags[dst]) return;
  int src = eidx[e];              // edge_index[0][e]
  atomicAdd(&agg[(size_t)dst * DIM + d], h[(size_t)src * DIM + d]);
  if (!FILTER && d == 0) atomicAdd(&deg[dst], 1.0f);
}

// ------------------------------------------------ layer 1: fused WMMA GEMM
// h1 = relu([mean1 | h0] @ [W_l1 ; W_r1] + b1), M=100000, K=128, N=64.
// One wave computes MT 16x16 tiles in M for one N tile; f32 WMMA 16x16x4.
// K loop split (mean half / h half) so control flow is wave-uniform.
__global__ void __launch_bounds__(32)
gemm1_kernel(const float* __restrict__ agg, const float* __restrict__ deg,
             const float* __restrict__ h0, const float* __restrict__ Wpk,
             const float* __restrict__ bias, float* __restrict__ h1,
             int nNTiles, int nNodes) {
  int tile  = blockIdx.x;
  int mg    = tile / nNTiles;
  int ntile = tile - mg * nNTiles;
  int lane  = threadIdx.x;
  int half  = lane >> 4;
  int l15   = lane & 15;
  int col   = ntile * 16 + l15;            // N index, < 64

  const float* aggRow[MT];
  const float* h0Row[MT];
  float invd[MT];
  for (int t = 0; t < MT; ++t) {
    int rowA = (mg * MT + t) * 16 + l15;
    int rA   = rowA < nNodes ? rowA : nNodes - 1;
    invd[t]  = 1.0f / fmaxf(deg[rA], 1.0f);
    aggRow[t] = agg + (size_t)rA * DIM;
    h0Row[t]  = h0  + (size_t)rA * DIM;
  }

  v8f c[MT];
  for (int t = 0; t < MT; ++t) c[t] = (v8f){0.f,0.f,0.f,0.f,0.f,0.f,0.f,0.f};

  // K = 0..63 : mean(agg*invd) x W_l
  for (int k0 = 0; k0 < 64; k0 += 4) {
    int ka = k0 + (half << 1);
    v2f b = *(const v2f*)(Wpk + ((size_t)((k0 >> 1) + half) * DIM + col) * 2);
    for (int t = 0; t < MT; ++t) {
      v2f raw = *(const v2f*)(aggRow[t] + ka);
      v2f a;
      a[0] = raw[0] * invd[t];
      a[1] = raw[1] * invd[t];
      c[t] = __builtin_amdgcn_wmma_f32_16x16x4_f32(false, a, false, b, (short)0,
                                                   c[t], false, false);
    }
  }
  // K = 64..127 : h0 x W_r
  for (int k0 = 64; k0 < 128; k0 += 4) {
    int ka = (k0 - 64) + (half << 1);
    v2f b = *(const v2f*)(Wpk + ((size_t)((k0 >> 1) + half) * DIM + col) * 2);
    for (int t = 0; t < MT; ++t) {
      v2f a = *(const v2f*)(h0Row[t] + ka);
      c[t] = __builtin_amdgcn_wmma_f32_16x16x4_f32(false, a, false, b, (short)0,
                                                   c[t], false, false);
    }
  }

  float bv = bias[col];
  for (int t = 0; t < MT; ++t) {
    int rowC0 = (mg * MT + t) * 16 + (half << 3);
    for (int v = 0; v < 8; ++v) {
      int row = rowC0 + v;
      if (row < nNodes)
        h1[(size_t)row * DIM + col] = fmaxf(c[t][v] + bv, 0.0f);
    }
  }
}

// --------------------------------- compact A for layer 2 (masked rows only)
__global__ void build_a2_kernel(const int* __restrict__ mask, const float* __restrict__ agg,
                                const float* __restrict__ deg, const float* __restrict__ h1,
                                float* __restrict__ A2, int nMask) {
  int gid = blockIdx.x * blockDim.x + threadIdx.x;
  int i = gid >> 7, cc = gid & 127;
  if (i >= nMask) return;
  int node = mask[i];
  float v;
  if (cc < 64) {
    float invd = 1.0f / fmaxf(deg[node], 1.0f);
    v = agg[(size_t)node * DIM + cc] * invd;
  } else {
    v = h1[(size_t)node * DIM + (cc - 64)];
  }
  A2[(size_t)i * K2 + cc] = v;
}

// ----------------------------------------------- layer 2: WMMA GEMM to out
// logits = A2(10000x128) @ Wpk2(packed [W_l2;W_r2], 128x2048) + b2
__global__ void __launch_bounds__(32)
gemm2_kernel(const float* __restrict__ A2, const float* __restrict__ Wpk,
             const float* __restrict__ bias, float* __restrict__ out,
             int nNTiles, int nMask, int vocab) {
  int tile  = blockIdx.x;
  int mg    = tile / nNTiles;
  int ntile = tile - mg * nNTiles;
  int lane  = threadIdx.x;
  int half  = lane >> 4;
  int l15   = lane & 15;
  int col   = ntile * 16 + l15;

  const float* aRow[MT];
  for (int t = 0; t < MT; ++t) {
    int rowA = (mg * MT + t) * 16 + l15;
    int rA   = rowA < nMask ? rowA : nMask - 1;
    aRow[t]  = A2 + (size_t)rA * K2;
  }

  v8f c[MT];
  for (int t = 0; t < MT; ++t) c[t] = (v8f){0.f,0.f,0.f,0.f,0.f,0.f,0.f,0.f};

  for (int k0 = 0; k0 < K2; k0 += 4) {
    int ka = k0 + (half << 1);
    v2f b = *(const v2f*)(Wpk + ((size_t)((k0 >> 1) + half) * vocab + col) * 2);
    for (int t = 0; t < MT; ++t) {
      v2f a = *(const v2f*)(aRow[t] + ka);
      c[t] = __builtin_amdgcn_wmma_f32_16x16x4_f32(false, a, false, b, (short)0,
                                                   c[t], false, false);
    }
  }

  float bv = bias[col];
  for (int t = 0; t < MT; ++t) {
    int rowC0 = (mg * MT + t) * 16 + (half << 3);
    for (int v = 0; v < 8; ++v) {
      int row = rowC0 + v;
      if (row < nMask)
        out[(size_t)row * vocab + col] = c[t][v] + bv;
    }
  }
}

// --------------------------------------------------- in-place log_softmax
__global__ void __launch_bounds__(256)
logsoftmax_kernel(float* __restrict__ out, int ncols) {
  __shared__ float red[256];
  int row = blockIdx.x;
  int tid = threadIdx.x;
  float* p = out + (size_t)row * ncols;

  float m = -INFINITY;
  for (int c = tid; c < ncols; c += 256) m = fmaxf(m, p[c]);
  red[tid] = m;
  __syncthreads();
  for (int s = 128; s > 0; s >>= 1) {
    if (tid < s) red[tid] = fmaxf(red[tid], red[tid + s]);
    __syncthreads();
  }
  float rowmax = red[0];
  __syncthreads();

  float sum = 0.0f;
  for (int c = tid; c < ncols; c += 256) sum += expf(p[c] - rowmax);
  red[tid] = sum;
  __syncthreads();
  for (int s = 128; s > 0; s >>= 1) {
    if (tid < s) red[tid] += red[tid + s];
    __syncthreads();
  }
  float lse = rowmax + logf(red[0]);
  __syncthreads();

  for (int c = tid; c < ncols; c += 256) p[c] = p[c] - lse;
}

// ------------------------------------------------------------------ driver
extern "C" void kernel_launch(void* const* d_in, const int* in_sizes, int n_in,
                              void* d_out, int out_size, void* d_ws, size_t ws_size,
                              hipStream_t stream) {
  const int*   x    = (const int*)d_in[0];
  const int*   eidx = (const int*)d_in[1];
  const int*   mask = (const int*)d_in[2];
  const float* emb  = (const float*)d_in[3];
  const float* Wl1  = (const float*)d_in[4];
  const float* b1   = (const float*)d_in[5];
  const float* Wr1  = (const float*)d_in[6];
  const float* Wl2  = (const float*)d_in[7];
  const float* b2   = (const float*)d_in[8];
  const float* Wr2  = (const float*)d_in[9];
  float* out = (float*)d_out;

  const int nNodes = in_sizes[0];
  const int nEdges = in_sizes[1] / 2;
  const int nMask  = in_sizes[2];
  const int vocab  = in_sizes[8];

  // ---- workspace carve (256B aligned)
  char* ws = (char*)d_ws;
  size_t off = 0;
  auto carve = [&](size_t bytes) -> void* {
    void* p = ws + off;
    off = (off + bytes + 255) & ~(size_t)255;
    return p;
  };
  float* h0    = (float*)carve((size_t)nNodes * DIM * sizeof(float));
  float* agg   = (float*)carve((size_t)nNodes * DIM * sizeof(float)); // reused L1 & L2
  float* h1    = (float*)carve((size_t)nNodes * DIM * sizeof(float));
  float* deg   = (float*)carve((size_t)nNodes * sizeof(float));
  int*   flags = (int*)  carve((size_t)nNodes * sizeof(int));
  float* A2    = (float*)carve((size_t)nMask * K2 * sizeof(float));
  float* Wpk1  = (float*)carve((size_t)K2 * DIM * sizeof(float));
  float* Wpk2  = (float*)carve((size_t)K2 * vocab * sizeof(float));
  (void)ws_size; (void)n_in; (void)out_size;

  const size_t aggN = (size_t)nNodes * DIM;

  // ---- weight packing (independent of graph data)
  pack_w_kernel<<<(K2 * DIM + 255) / 256, 256, 0, stream>>>(Wl1, Wr1, Wpk1, DIM);
  pack_w_kernel<<<(K2 * vocab + 255) / 256, 256, 0, stream>>>(Wl2, Wr2, Wpk2, vocab);

  // ---- layer 1
  zero_f32_kernel<<<2048, 256, 0, stream>>>(agg, aggN);
  zero_f32_kernel<<<512, 256, 0, stream>>>(deg, (size_t)nNodes);
  zero_f32_kernel<<<512, 256, 0, stream>>>((float*)flags, (size_t)nNodes);

  {
    long long tot = (long long)nNodes * DIM;
    gather_kernel<<<(unsigned)((tot + 255) / 256), 256, 0, stream>>>(x, emb, h0, nNodes);
  }
  {
    long long tot = (long long)nEdges * DIM;
    scatter_kernel<false><<<(unsigned)((tot + 255) / 256), 256, 0, stream>>>(
        eidx, h0, agg, deg, nullptr, nEdges);
  }
  {
    int mtiles  = (nNodes + 15) / 16;          // 6250
    int mgroups = (mtiles + MT - 1) / MT;      // 1563
    int ntiles  = DIM / 16;                    // 4
    gemm1_kernel<<<mgroups * ntiles, 32, 0, stream>>>(agg, deg, h0, Wpk1, b1, h1,
                                                      ntiles, nNodes);
  }

  // ---- layer 2 (only masked destinations matter)
  zero_f32_kernel<<<2048, 256, 0, stream>>>(agg, aggN);
  flags_kernel<<<(nMask + 255) / 256, 256, 0, stream>>>(mask, flags, nMask);
  {
    long long tot = (long long)nEdges * DIM;
    scatter_kernel<true><<<(unsigned)((tot + 255) / 256), 256, 0, stream>>>(
        eidx, h1, agg, nullptr, flags, nEdges);
  }
  {
    long long tot = (long long)nMask * K2;
    build_a2_kernel<<<(unsigned)((tot + 255) / 256), 256, 0, stream>>>(
        mask, agg, deg, h1, A2, nMask);
  }
  {
    int mtiles  = (nMask + 15) / 16;           // 625
    int mgroups = (mtiles + MT - 1) / MT;      // 157
    int ntiles  = vocab / 16;                  // 128
    gemm2_kernel<<<mgroups * ntiles, 32, 0, stream>>>(A2, Wpk2, b2, out,
                                                      ntiles, nMask, vocab);
  }
  logsoftmax_kernel<<<nMask, 256, 0, stream>>>(out, vocab);
}